// AtomDistances_41051297415622
// MI455X (gfx1250) — compile-verified
//
#include <hip/hip_runtime.h>
#include <stdint.h>

// Problem constants from the reference: B=64, N=512, NB=256.
#define B_DIM    64
#define N_DIM    512
#define NB_DIM   256
#define ROWS     16      // rows of the [N, NB] output handled per block
#define THREADS  256     // 8 wave32s

// Native clang vectors (HIP's int4/float4 are wrapper classes that
// __builtin_nontemporal_load/store rejects).
typedef int   v4i __attribute__((ext_vector_type(4)));
typedef float v4f __attribute__((ext_vector_type(4)));

// distances[b,i,nb] = mask ? || pos[b, nbr[b,i,nb]] - pos[b,i] || : 0
//
// Memory-bound (~100 MB HBM traffic -> ~4.3 us floor @ 23.3 TB/s).
// - Per-batch position table (6 KB) staged into LDS via gfx1250 async
//   global->LDS DMA (ASYNCcnt), padded to 16-B slots so each random
//   gather is a single aligned ds_load_b128.
// - Index/mask/output streams are touched exactly once: 4-wide b128
//   accesses with non-temporal cache hints.
// - __builtin_amdgcn_sqrtf -> bare v_sqrt_f32 (no IEEE fixup sequence).
__global__ __launch_bounds__(THREADS) void atom_distances_kernel(
    const float* __restrict__ positions,      // [B, N, 3] f32
    const int*   __restrict__ neighbors,      // [B, N, NB] indices (int32 per harness)
    const int*   __restrict__ neighbor_mask,  // [B, N, NB] 0/1
    float*       __restrict__ out)            // [B, N, NB] f32
{
    __shared__ v4f s_pos[N_DIM];              // 8 KB of the 320 KB WGP LDS

    const int tid = threadIdx.x;
    const int blocks_per_batch = N_DIM / ROWS;                 // 32
    const int b  = blockIdx.x / blocks_per_batch;
    const int i0 = (blockIdx.x % blocks_per_batch) * ROWS;

    // ---- Stage this batch's positions (512 atoms x 12 B packed xyz) into LDS
    // using gfx1250 async global->LDS loads: xy as b64, z as b32, into 16 B slots.
    {
        const char*    gbase = (const char*)(positions + (size_t)b * N_DIM * 3);
        const unsigned lbase = (unsigned)(uintptr_t)s_pos;  // low 32 bits of flat
                                                            // shared ptr == LDS byte offset
#pragma unroll
        for (int k = 0; k < N_DIM / THREADS; ++k) {         // 2 atoms per thread
            const int      atom = tid + k * THREADS;
            const uint64_t g    = (uint64_t)(uintptr_t)(gbase + atom * 12);
            const unsigned l    = lbase + (unsigned)atom * 16u;
            asm volatile("global_load_async_to_lds_b64 %0, %1, off"
                         :: "v"(l), "v"(g) : "memory");
            asm volatile("global_load_async_to_lds_b32 %0, %1, off"
                         :: "v"(l + 8u), "v"(g + 8) : "memory");
        }
        // Wait for this wave's async DMA into LDS to land.
        asm volatile("s_wait_asynccnt 0x0" ::: "memory");
    }
    __syncthreads();  // other waves' staging must also be visible

    // ---- Streaming phase: 4 neighbor slots per thread, b128 everywhere.
    // 64 threads cover one row (64 x 4 = 256 cols); 256 threads = 4 rows/iter.
    const int rsel = tid >> 6;          // row within the 4-row group
    const int c4   = (tid & 63) * 4;    // starting column (16 B aligned)

#pragma unroll
    for (int it = 0; it < ROWS / 4; ++it) {
        const int i  = i0 + it * 4 + rsel;
        const v4f pi = s_pos[i];                                   // broadcast ds_load_b128

        const size_t off = ((size_t)b * N_DIM + i) * NB_DIM + c4;
        const v4i jj = __builtin_nontemporal_load((const v4i*)(neighbors     + off));
        const v4i mm = __builtin_nontemporal_load((const v4i*)(neighbor_mask + off));

        const v4f p0 = s_pos[jj.x & (N_DIM - 1)];                  // random ds_load_b128 x4
        const v4f p1 = s_pos[jj.y & (N_DIM - 1)];
        const v4f p2 = s_pos[jj.z & (N_DIM - 1)];
        const v4f p3 = s_pos[jj.w & (N_DIM - 1)];

        v4f r;
        {
            const float dx = p0.x - pi.x, dy = p0.y - pi.y, dz = p0.z - pi.z;
            r.x = __builtin_amdgcn_sqrtf(__builtin_fmaf(dx, dx, __builtin_fmaf(dy, dy, dz * dz)));
        }
        {
            const float dx = p1.x - pi.x, dy = p1.y - pi.y, dz = p1.z - pi.z;
            r.y = __builtin_amdgcn_sqrtf(__builtin_fmaf(dx, dx, __builtin_fmaf(dy, dy, dz * dz)));
        }
        {
            const float dx = p2.x - pi.x, dy = p2.y - pi.y, dz = p2.z - pi.z;
            r.z = __builtin_amdgcn_sqrtf(__builtin_fmaf(dx, dx, __builtin_fmaf(dy, dy, dz * dz)));
        }
        {
            const float dx = p3.x - pi.x, dy = p3.y - pi.y, dz = p3.z - pi.z;
            r.w = __builtin_amdgcn_sqrtf(__builtin_fmaf(dx, dx, __builtin_fmaf(dy, dy, dz * dz)));
        }
        r.x = mm.x ? r.x : 0.0f;
        r.y = mm.y ? r.y : 0.0f;
        r.z = mm.z ? r.z : 0.0f;
        r.w = mm.w ? r.w : 0.0f;

        __builtin_nontemporal_store(r, (v4f*)(out + off));         // b128 NT store
    }
}

extern "C" void kernel_launch(void* const* d_in, const int* in_sizes, int n_in,
                              void* d_out, int out_size, void* d_ws, size_t ws_size,
                              hipStream_t stream)
{
    (void)in_sizes; (void)n_in; (void)d_ws; (void)ws_size; (void)out_size;

    const float* positions     = (const float*)d_in[0];
    const int*   neighbors     = (const int*)d_in[1];
    const int*   neighbor_mask = (const int*)d_in[2];
    float*       out           = (float*)d_out;

    dim3 grid(B_DIM * (N_DIM / ROWS));   // 2048 blocks
    dim3 block(THREADS);                 // 256 threads = 8 wave32s
    hipLaunchKernelGGL(atom_distances_kernel, grid, block, 0, stream,
                       positions, neighbors, neighbor_mask, out);
}